// GAT_43404939494166
// MI455X (gfx1250) — compile-verified
//
#include <hip/hip_runtime.h>
#include <hip/hip_bf16.h>

#define N_NODES   50000
#define N_EDGES   800000
#define IN_DIM    256
#define HID       64
#define HEADS     4
#define HD        256        // HEADS*HID
#define N_CLASSES 40
#define NEG_SLOPE 0.2f

typedef __attribute__((ext_vector_type(2))) float v2f;
typedef __attribute__((ext_vector_type(8))) float v8f;

#define WMMA_F32(a, b, c) \
    __builtin_amdgcn_wmma_f32_16x16x4_f32(false, (a), false, (b), (short)0, (c), false, false)

// ---------------------------------------------------------------------------
// Kernel 1: H = features @ W0   ([N,256] x [256,256] -> [N,256])
// One wave per 16x64 strip: 4 N-tiles share one A fragment per K-step
// (4x less A traffic). 50000 = 3125*16 rows exactly; K=256 exact.
// ---------------------------------------------------------------------------
__global__ __launch_bounds__(256) void gat_gemm_h(const float* __restrict__ A,
                                                  const float* __restrict__ B,
                                                  float* __restrict__ H) {
    const int lane = threadIdx.x & 31;
    const int wid  = blockIdx.x * 8 + (threadIdx.x >> 5);
    if (wid >= 3125 * 4) return;          // wave-uniform: skips WMMA entirely
    const int mt   = wid >> 2;            // 3125 M tiles (16 rows each)
    const int ng   = wid & 3;             // 4 groups of 64 columns
    const int row  = lane & 15;
    const int koff = (lane >> 4) << 1;    // lanes 0-15 -> K,K+1 ; 16-31 -> K+2,K+3
    const int col  = lane & 15;

    const float* arow  = A + (size_t)(mt * 16 + row) * IN_DIM;
    const float* bbase = B + ng * 64 + col;

    v8f c0 = {}, c1 = {}, c2 = {}, c3 = {};
    for (int k = 0; k < IN_DIM; k += 4) {
        v2f a;
        a.x = arow[k + koff];
        a.y = arow[k + koff + 1];
        const float* b0 = bbase + (size_t)(k + koff)     * HD;
        const float* b1 = bbase + (size_t)(k + koff + 1) * HD;
        v2f b;
        b.x = b0[0];  b.y = b1[0];  c0 = WMMA_F32(a, b, c0);
        b.x = b0[16]; b.y = b1[16]; c1 = WMMA_F32(a, b, c1);
        b.x = b0[32]; b.y = b1[32]; c2 = WMMA_F32(a, b, c2);
        b.x = b0[48]; b.y = b1[48]; c3 = WMMA_F32(a, b, c3);
    }
    const int m0 = mt * 16 + ((lane >> 4) << 3);   // lanes 16-31 hold rows M=8..15
    float* out = H + (size_t)m0 * HD + ng * 64 + col;
#pragma unroll
    for (int i = 0; i < 8; ++i) {
        float* o = out + (size_t)i * HD;
        o[0]  = c0[i];
        o[16] = c1[i];
        o[32] = c2[i];
        o[48] = c3[i];
    }
}

// ---------------------------------------------------------------------------
// Kernel 2: el/er = einsum('nhd,hd->nh').  One wave per (node, head):
// 32 lanes x float2 covers HID=64, shfl_xor tree reduction.
// ---------------------------------------------------------------------------
__global__ __launch_bounds__(256) void gat_attn_coef(const float* __restrict__ H,
                                                     const float* __restrict__ al,
                                                     const float* __restrict__ ar,
                                                     float* __restrict__ el,
                                                     float* __restrict__ er) {
    const int lane = threadIdx.x & 31;
    const int w    = blockIdx.x * 8 + (threadIdx.x >> 5);   // node*HEADS + head
    const int node = w >> 2;
    const int head = w & 3;
    const int base = head * HID + lane * 2;

    const float* hp = H + (size_t)node * HD + base;
    const float h0 = hp[0], h1 = hp[1];
    float pl = h0 * al[base] + h1 * al[base + 1];
    float pr = h0 * ar[base] + h1 * ar[base + 1];
#pragma unroll
    for (int off = 16; off > 0; off >>= 1) {
        pl += __shfl_xor(pl, off, 32);
        pr += __shfl_xor(pr, off, 32);
    }
    if (lane == 0) {
        el[node * HEADS + head] = pl;
        er[node * HEADS + head] = pr;
    }
}

// ---------------------------------------------------------------------------
// Kernel 3: per (edge, head): e = leakyrelu(el[src]+er[dst]); segment-max into
// smax[dst] via the race-safe int-max / uint-min float trick (smax preset 0xFF).
// ---------------------------------------------------------------------------
__global__ __launch_bounds__(256) void gat_edge_max(const int* __restrict__ src,
                                                    const int* __restrict__ dst,
                                                    const float* __restrict__ el,
                                                    const float* __restrict__ er,
                                                    float* __restrict__ ebuf,
                                                    float* __restrict__ smax) {
    const int t = blockIdx.x * blockDim.x + threadIdx.x;
    if (t >= N_EDGES * HEADS) return;
    const int e = t >> 2, h = t & 3;
    const int s = src[e], d = dst[e];
    float v = el[s * HEADS + h] + er[d * HEADS + h];
    v = v > 0.f ? v : NEG_SLOPE * v;
    ebuf[t] = v;
    const int idx = d * HEADS + h;
    if (v >= 0.f) atomicMax((int*)(smax + idx), __float_as_int(v));
    else          atomicMin((unsigned int*)(smax + idx), __float_as_uint(v));
}

// ---------------------------------------------------------------------------
// Kernel 4: ex = exp(e - smax[dst]); ssum[dst] += ex; ebuf <- ex.
// ---------------------------------------------------------------------------
__global__ __launch_bounds__(256) void gat_edge_exp(const int* __restrict__ dst,
                                                    float* __restrict__ ebuf,
                                                    const float* __restrict__ smax,
                                                    float* __restrict__ ssum) {
    const int t = blockIdx.x * blockDim.x + threadIdx.x;
    if (t >= N_EDGES * HEADS) return;
    const int e = t >> 2, h = t & 3;
    const int idx = dst[e] * HEADS + h;
    const float ex = __expf(ebuf[t] - smax[idx]);
    ebuf[t] = ex;
    atomicAdd(ssum + idx, ex);
}

// ---------------------------------------------------------------------------
// Kernel 5: one wave per edge. lane -> 8 dims (head = lane/8).
// alpha = ex / ssum[dst]; h_out[dst] += alpha * h[src]  (f32 atomics, L2-resident)
// ---------------------------------------------------------------------------
__global__ __launch_bounds__(256) void gat_edge_agg(const int* __restrict__ src,
                                                    const int* __restrict__ dst,
                                                    const float* __restrict__ ebuf,
                                                    const float* __restrict__ ssum,
                                                    const float* __restrict__ H,
                                                    float* __restrict__ HO) {
    const int lane = threadIdx.x & 31;
    const int e    = blockIdx.x * 8 + (threadIdx.x >> 5);
    const int s = src[e], d = dst[e];
    const int h = lane >> 3;
    const float alpha = ebuf[e * HEADS + h] / ssum[d * HEADS + h];

    const float4* hp = (const float4*)(H + (size_t)s * HD + lane * 8);
    const float4 x0 = hp[0], x1 = hp[1];
    float* op = HO + (size_t)d * HD + lane * 8;
    atomicAdd(op + 0, alpha * x0.x);
    atomicAdd(op + 1, alpha * x0.y);
    atomicAdd(op + 2, alpha * x0.z);
    atomicAdd(op + 3, alpha * x0.w);
    atomicAdd(op + 4, alpha * x1.x);
    atomicAdd(op + 5, alpha * x1.y);
    atomicAdd(op + 6, alpha * x1.z);
    atomicAdd(op + 7, alpha * x1.w);
}

// ---------------------------------------------------------------------------
// Kernel 6: OUT = relu(HO) @ fc_W + fc_b  ([N,256] x [256,40]).
// One wave per 16-row block computes all 3 N-tiles (HO read once). ReLU fused
// into the shared A fragment. Tile 2 (cols 32..47) is multiply-masked so EXEC
// stays all-ones through the WMMAs; divergence only at the final store.
// ---------------------------------------------------------------------------
__global__ __launch_bounds__(256) void gat_gemm_out(const float* __restrict__ HO,
                                                    const float* __restrict__ W,
                                                    const float* __restrict__ bias,
                                                    float* __restrict__ OUT) {
    const int lane = threadIdx.x & 31;
    const int mt   = blockIdx.x * 8 + (threadIdx.x >> 5);
    if (mt >= 3125) return;                   // wave-uniform: skips WMMA entirely

    const int row  = lane & 15;
    const int koff = (lane >> 4) << 1;
    const int col  = lane & 15;
    const int c2col = 32 + col;
    const float m2  = (c2col < N_CLASSES) ? 1.f : 0.f;
    const int   c2i = (c2col < N_CLASSES) ? c2col : 0;

    const float* arow = HO + (size_t)(mt * 16 + row) * HD;
    v8f c0 = {}, c1 = {}, c2 = {};
    for (int k = 0; k < HD; k += 4) {
        v2f a;
        a.x = fmaxf(arow[k + koff], 0.f);                 // fused ReLU
        a.y = fmaxf(arow[k + koff + 1], 0.f);
        const float* w0 = W + (size_t)(k + koff)     * N_CLASSES;
        const float* w1 = W + (size_t)(k + koff + 1) * N_CLASSES;
        v2f b;
        b.x = w0[col];           b.y = w1[col];           c0 = WMMA_F32(a, b, c0);
        b.x = w0[16 + col];      b.y = w1[16 + col];      c1 = WMMA_F32(a, b, c1);
        b.x = w0[c2i] * m2;      b.y = w1[c2i] * m2;      c2 = WMMA_F32(a, b, c2);
    }
    const int m0 = mt * 16 + ((lane >> 4) << 3);
    const float bv0 = bias[col];
    const float bv1 = bias[16 + col];
    const float bv2 = (c2col < N_CLASSES) ? bias[c2col] : 0.f;
#pragma unroll
    for (int i = 0; i < 8; ++i) {
        float* o = OUT + (size_t)(m0 + i) * N_CLASSES;
        o[col]      = c0[i] + bv0;
        o[16 + col] = c1[i] + bv1;
        if (c2col < N_CLASSES) o[c2col] = c2[i] + bv2;
    }
}

// ---------------------------------------------------------------------------
extern "C" void kernel_launch(void* const* d_in, const int* in_sizes, int n_in,
                              void* d_out, int out_size, void* d_ws, size_t ws_size,
                              hipStream_t stream) {
    const float* features = (const float*)d_in[0];   // [N, 256]
    const float* W0       = (const float*)d_in[1];   // [256, 256]
    const float* attn_l   = (const float*)d_in[2];   // [4, 64]
    const float* attn_r   = (const float*)d_in[3];   // [4, 64]
    const float* fc_W     = (const float*)d_in[4];   // [256, 40]
    const float* fc_b     = (const float*)d_in[5];   // [40]
    const int*   src      = (const int*)d_in[6];     // [E]
    const int*   dst      = (const int*)d_in[7];     // [E]
    float*       out      = (float*)d_out;           // [N, 40]

    // Workspace carve-up (floats): ~118 MB total
    float* H    = (float*)d_ws;                          // [N,256] projected feats
    float* HO   = H    + (size_t)N_NODES * HD;           // [N,256] aggregated
    float* el   = HO   + (size_t)N_NODES * HD;           // [N,4]
    float* er   = el   + (size_t)N_NODES * HEADS;        // [N,4]
    float* smax = er   + (size_t)N_NODES * HEADS;        // [N,4]
    float* ssum = smax + (size_t)N_NODES * HEADS;        // [N,4]
    float* ebuf = ssum + (size_t)N_NODES * HEADS;        // [E,4]

    // Inits (graph-capture-safe async memsets).
    hipMemsetAsync(smax, 0xFF, (size_t)N_NODES * HEADS * sizeof(float), stream); // -inf-like sentinel
    hipMemsetAsync(ssum, 0x00, (size_t)N_NODES * HEADS * sizeof(float), stream);
    hipMemsetAsync(HO,   0x00, (size_t)N_NODES * HD    * sizeof(float), stream);

    // 1) H = features @ W0 : 3125 M-tiles x 4 col-groups = 12500 waves
    gat_gemm_h<<<1563, 256, 0, stream>>>(features, W0, H);

    // 2) el/er : N*HEADS = 200000 waves
    gat_attn_coef<<<25000, 256, 0, stream>>>(H, attn_l, attn_r, el, er);

    // 3) edge leaky-relu + segment max : E*HEADS = 3.2M threads
    gat_edge_max<<<12500, 256, 0, stream>>>(src, dst, el, er, ebuf, smax);

    // 4) edge exp + segment sum
    gat_edge_exp<<<12500, 256, 0, stream>>>(dst, ebuf, smax, ssum);

    // 5) weighted aggregation : one wave per edge, 8 edges/block
    gat_edge_agg<<<100000, 256, 0, stream>>>(src, dst, ebuf, ssum, H, HO);

    // 6) OUT = relu(HO) @ fc_W + fc_b : 3125 waves, each does all 3 N-tiles
    gat_gemm_out<<<391, 256, 0, stream>>>(HO, fc_W, fc_b, out);
}